// GCN_90709709292172
// MI455X (gfx1250) — compile-verified
//
#include <hip/hip_runtime.h>
#include <hip/hip_bf16.h>

// ---------------------------------------------------------------------------
// GCN on MI455X (gfx1250). fp32 WMMA 16x16x4 for dense transforms; fused
// epilogues (self-loop+bias prefill) and fused ReLU-on-load to minimize
// memory passes. Scatter is float4-vectorized atomic adds (L2-resident).
// ---------------------------------------------------------------------------

typedef float v2f __attribute__((ext_vector_type(2)));
typedef float v8f __attribute__((ext_vector_type(8)));

#define D_HID 64

// ---------------- degree / normalization ----------------

__global__ void k_fill(float* __restrict__ p, float v, int n) {
    int i = blockIdx.x * blockDim.x + threadIdx.x;
    if (i < n) p[i] = v;
}

__global__ void k_deg(const int* __restrict__ dst, float* __restrict__ deg, int E) {
    int e = blockIdx.x * blockDim.x + threadIdx.x;
    if (e < E) atomicAdd(&deg[dst[e]], 1.0f);
}

__global__ void k_rsqrt(float* __restrict__ d, int n) {
    int i = blockIdx.x * blockDim.x + threadIdx.x;
    if (i < n) d[i] = 1.0f / sqrtf(d[i]);
}

// ---------------- edge scatter (float4 vectorized) ----------------
// agg[dst, f:f+4] += h[src, f:f+4] * dinv[src]*dinv[dst]
__global__ void k_scatter(const int* __restrict__ src, const int* __restrict__ dst,
                          const float* __restrict__ dinv, const float* __restrict__ h,
                          float* __restrict__ agg, int E) {
    int gid = blockIdx.x * blockDim.x + threadIdx.x;
    int e = gid >> 4;                 // 16 threads per edge
    int f = (gid & 15) * 4;           // 4 features per thread
    if (e < E) {
        int s = src[e];
        int d = dst[e];
        float nrm = dinv[s] * dinv[d];
        const float4 hv = *(const float4*)(h + (size_t)s * D_HID + f);
        float* ap = agg + (size_t)d * D_HID + f;
        atomicAdd(ap + 0, hv.x * nrm);
        atomicAdd(ap + 1, hv.y * nrm);
        atomicAdd(ap + 2, hv.z * nrm);
        atomicAdd(ap + 3, hv.w * nrm);
    }
}

// ---------------- fp32 WMMA GEMM with fused epilogues ----------------
//
// Y[nrows x (NT*16)] = act(X)[nrows x 64] @ W  (+ bias if BIAS)
// If PRE:  also write AGG = Y * dinv[row]^2 + b[col]  (self-loop + bias
//          prefill for the subsequent edge scatter).
// If RELU_IN: A-matrix elements pass through fmaxf(.,0) on load (applies
//          the previous layer's activation without an extra memory pass).
//
// One wave per 16-row tile; K = 64 stepped by 4 with V_WMMA_F32_16X16X4_F32.
// VGPR layouts per CDNA5 ISA 7.12.2:
//   A 16x4:  lane (l&15)=M, half=(l>>4): VGPR v holds K = v + 2*half
//   B 4x16:  lane (l&15)=N, VGPR v holds K = v + 2*half
//   C/D:     VGPR r: lanes 0-15 -> M=r, lanes 16-31 -> M=r+8, N=(l&15)
template <int NT, bool RELU_IN, bool PRE, bool BIAS>
__global__ __launch_bounds__(128) void k_gemm(const float* __restrict__ X,
                                              const float* __restrict__ W,
                                              const float* __restrict__ bias,
                                              const float* __restrict__ dinv,
                                              float* __restrict__ Y,
                                              float* __restrict__ agg,
                                              int nrows) {
    const int ncols = NT * 16;
    int wave = threadIdx.x >> 5;
    int lane = threadIdx.x & 31;
    int rb = blockIdx.x * 4 + wave;          // row-tile id (uniform per wave)
    if (rb * 16 >= nrows) return;            // wave-uniform exit: EXEC all-1s at WMMA
    int m0 = rb * 16;
    int half = lane >> 4;                    // 0 or 1
    int ml = lane & 15;

    v8f acc[NT];
#pragma unroll
    for (int t = 0; t < NT; ++t)
        acc[t] = (v8f){0.f, 0.f, 0.f, 0.f, 0.f, 0.f, 0.f, 0.f};

    const float* arow = X + (size_t)(m0 + ml) * D_HID;

#pragma unroll
    for (int kk = 0; kk < D_HID; kk += 4) {
        int k0 = kk + 2 * half;              // even -> 8B aligned float2 load
        v2f a = *(const v2f*)(arow + k0);
        if (RELU_IN) {
            a.x = fmaxf(a.x, 0.0f);
            a.y = fmaxf(a.y, 0.0f);
        }
#pragma unroll
        for (int t = 0; t < NT; ++t) {
            v2f b;
            b.x = W[(size_t)k0 * ncols + t * 16 + ml];
            b.y = W[(size_t)(k0 + 1) * ncols + t * 16 + ml];
            acc[t] = __builtin_amdgcn_wmma_f32_16x16x4_f32(
                false, a, false, b, (short)0, acc[t], false, false);
        }
    }

#pragma unroll
    for (int t = 0; t < NT; ++t) {
        int col = t * 16 + ml;
        float bb = (PRE || BIAS) ? bias[col] : 0.0f;
#pragma unroll
        for (int r = 0; r < 8; ++r) {
            int row = m0 + r + 8 * half;
            float v = acc[t][r];
            size_t idx = (size_t)row * ncols + col;
            Y[idx] = BIAS ? (v + bb) : v;
            if (PRE) {
                float di = dinv[row];
                agg[idx] = v * di * di + bb;  // self-loop message + bias prefill
            }
        }
    }
}

// ---------------- host launcher ----------------

static inline int cdiv_i(long long a, long long b) { return (int)((a + b - 1) / b); }

extern "C" void kernel_launch(void* const* d_in, const int* in_sizes, int n_in,
                              void* d_out, int out_size, void* d_ws, size_t ws_size,
                              hipStream_t stream) {
    const float* x  = (const float*)d_in[0];
    const int*   ei = (const int*)d_in[1];   // [2, E] int32
    const float* W1 = (const float*)d_in[2];
    const float* b1 = (const float*)d_in[3];
    const float* W2 = (const float*)d_in[4];
    const float* b2 = (const float*)d_in[5];
    const float* Wl = (const float*)d_in[6];
    const float* bl = (const float*)d_in[7];
    float* out = (float*)d_out;

    const int N = in_sizes[0] / D_HID;       // 100000
    const int E = in_sizes[1] / 2;           // 1200000
    const int* src = ei;                     // edge_index[0] = message sources
    const int* dst = ei + E;                 // edge_index[1] = aggregation targets

    // workspace (floats): dinv[Np] | bufH[Np*64] | agg1[Np*64] | agg2[Np*64]
    float* ws = (float*)d_ws;
    size_t Np = ((size_t)N + 63) & ~(size_t)63;
    float* dinv = ws;
    float* bufH = dinv + Np;                 // transformed features h = act(X) @ W
    float* agg1 = bufH + Np * D_HID;         // layer-1 aggregation accumulator
    float* agg2 = agg1 + Np * D_HID;         // layer-2 aggregation accumulator

    const int T = 256;
    const int gEF4  = cdiv_i((long long)E * (D_HID / 4), T);  // edge scatter grid
    const int gGemm = cdiv_i((N + 15) / 16, 4);               // 4 waves / block

    // --- symmetric normalization: deg = 1 (self loop) + in-degree; dinv = deg^-1/2
    k_fill<<<cdiv_i(N, T), T, 0, stream>>>(dinv, 1.0f, N);
    k_deg<<<cdiv_i(E, T), T, 0, stream>>>(dst, dinv, E);
    k_rsqrt<<<cdiv_i(N, T), T, 0, stream>>>(dinv, N);

    // --- layer 1: h1 = x@W1; agg1 = h1*dinv^2 + b1; scatter edges into agg1
    k_gemm<4, false, true, false><<<gGemm, 128, 0, stream>>>(
        x, W1, b1, dinv, bufH, agg1, N);
    k_scatter<<<gEF4, T, 0, stream>>>(src, dst, dinv, bufH, agg1, E);

    // --- layer 2: h2 = relu(agg1)@W2; agg2 = h2*dinv^2 + b2; scatter
    k_gemm<4, true, true, false><<<gGemm, 128, 0, stream>>>(
        agg1, W2, b2, dinv, bufH, agg2, N);
    k_scatter<<<gEF4, T, 0, stream>>>(src, dst, dinv, bufH, agg2, E);

    // --- head: out = relu(agg2) @ Wl + bl   (100K x 32)
    k_gemm<2, true, false, true><<<gGemm, 128, 0, stream>>>(
        agg2, Wl, bl, nullptr, out, nullptr, N);
}